// PyGAT_32950989095028
// MI455X (gfx1250) — compile-verified
//
#include <hip/hip_runtime.h>
#include <math.h>

typedef float v2f __attribute__((ext_vector_type(2)));
typedef float v8f __attribute__((ext_vector_type(8)));

#define FIN 512
#define HID 64
#define NCLS 16
#define NEG_SLOPE 0.2f

// ---------------- float atomic max via int punning (mixed-sign safe) --------
__device__ __forceinline__ void atomicMaxF32(float* addr, float val) {
  if (val >= 0.0f) {
    atomicMax((int*)addr, __float_as_int(val));
  } else {
    atomicMin((unsigned int*)addr, __float_as_uint(val));
  }
}

// ---------------- GEMM1: h1[N,64] = x[N,512] @ W1[512,64] via f32 WMMA ------
// One wave per 16-row tile. A fragment reused across 4 column tiles (cols 0-63),
// so x is streamed exactly once. W1 (128 KB) stays hot in cache.
__global__ void __launch_bounds__(128) gemm1_wmma(
    const float* __restrict__ x, const float* __restrict__ W1,
    float* __restrict__ h1, int nrows) {
  const int lane = threadIdx.x & 31;
  const int wave = threadIdx.x >> 5;
  const int m0 = (blockIdx.x * 4 + wave) * 16;
  if (m0 >= nrows) return;            // wave-uniform: EXEC stays all-ones for WMMA
  const int hi = lane >> 4;           // 0: K={0,1}, 1: K={2,3}
  const int n  = lane & 15;           // row (A) / col (B,C,D) within tile

  const float* xrow = x + (size_t)(m0 + n) * FIN + 2 * hi;
  v8f acc0 = {}, acc1 = {}, acc2 = {}, acc3 = {};

  for (int k = 0; k < FIN; k += 4) {
    float2 av = *(const float2*)(xrow + k);     // x[m, k+2*hi .. +1]
    v2f a; a.x = av.x; a.y = av.y;
    const float* wk = W1 + (size_t)(k + 2 * hi) * HID + n;
    v2f b0, b1v, b2v, b3v;
    b0.x  = wk[0];   b0.y  = wk[HID];           // B[K, n+0],  B[K+1, n+0]
    b1v.x = wk[16];  b1v.y = wk[HID + 16];
    b2v.x = wk[32];  b2v.y = wk[HID + 32];
    b3v.x = wk[48];  b3v.y = wk[HID + 48];
    acc0 = __builtin_amdgcn_wmma_f32_16x16x4_f32(false, a, false, b0,  (short)0, acc0, false, false);
    acc1 = __builtin_amdgcn_wmma_f32_16x16x4_f32(false, a, false, b1v, (short)0, acc1, false, false);
    acc2 = __builtin_amdgcn_wmma_f32_16x16x4_f32(false, a, false, b2v, (short)0, acc2, false, false);
    acc3 = __builtin_amdgcn_wmma_f32_16x16x4_f32(false, a, false, b3v, (short)0, acc3, false, false);
  }
  // C/D layout: vgpr r holds row m0 + r + 8*hi, col = (lane&15) + 16*tile
  float* orow = h1 + (size_t)(m0 + 8 * hi) * HID + n;
#pragma unroll
  for (int r = 0; r < 8; ++r) {
    orow[(size_t)r * HID +  0] = acc0[r];
    orow[(size_t)r * HID + 16] = acc1[r];
    orow[(size_t)r * HID + 32] = acc2[r];
    orow[(size_t)r * HID + 48] = acc3[r];
  }
}

// ---------------- per-(node,head) attention logits for layer 1 --------------
__global__ void node_attn1(const float* __restrict__ h1,
                           const float* __restrict__ asrc,
                           const float* __restrict__ adst,
                           float* __restrict__ es, float* __restrict__ ed, int nH) {
  int i = blockIdx.x * blockDim.x + threadIdx.x;   // i = node*8 + head
  if (i >= nH) return;
  int h = i & 7;
  const float* hp = h1 + (size_t)i * 8;
  float s = 0.f, d = 0.f;
#pragma unroll
  for (int c = 0; c < 8; ++c) {
    float v = hp[c];
    s += v * asrc[h * 8 + c];
    d += v * adst[h * 8 + c];
  }
  es[i] = s; ed[i] = d;
}

// ---------------- fill -------------------------------------------------------
__global__ void fillf(float* __restrict__ p, float v, size_t n) {
  size_t i = (size_t)blockIdx.x * blockDim.x + threadIdx.x;
  if (i < n) p[i] = v;
}

// ---------------- edge passes (thread per edge*head) ------------------------
template <int LOGH>
__global__ void edge_max_k(const int* __restrict__ src, const int* __restrict__ dst,
                           int E, int Etot,
                           const float* __restrict__ es, const float* __restrict__ ed,
                           float* __restrict__ mx) {
  int t = blockIdx.x * blockDim.x + threadIdx.x;
  int i = t >> LOGH, h = t & ((1 << LOGH) - 1);
  if (i >= Etot) return;
  int s = (i < E) ? src[i] : (i - E);
  int d = (i < E) ? dst[i] : (i - E);
  const int H = 1 << LOGH;
  float e = es[s * H + h] + ed[d * H + h];
  e = e > 0.f ? e : NEG_SLOPE * e;
  atomicMaxF32(&mx[d * H + h], e);
}

template <int LOGH>
__global__ void edge_sum_k(const int* __restrict__ src, const int* __restrict__ dst,
                           int E, int Etot,
                           const float* __restrict__ es, const float* __restrict__ ed,
                           const float* __restrict__ mx, float* __restrict__ ssum) {
  int t = blockIdx.x * blockDim.x + threadIdx.x;
  int i = t >> LOGH, h = t & ((1 << LOGH) - 1);
  if (i >= Etot) return;
  int s = (i < E) ? src[i] : (i - E);
  int d = (i < E) ? dst[i] : (i - E);
  const int H = 1 << LOGH;
  float e = es[s * H + h] + ed[d * H + h];
  e = e > 0.f ? e : NEG_SLOPE * e;
  atomicAdd(&ssum[d * H + h], __expf(e - mx[d * H + h]));
}

template <int LOGH, int C>
__global__ void edge_aggr_k(const int* __restrict__ src, const int* __restrict__ dst,
                            int E, int Etot,
                            const float* __restrict__ es, const float* __restrict__ ed,
                            const float* __restrict__ mx, const float* __restrict__ ssum,
                            const float* __restrict__ hsrc, float* __restrict__ agg) {
  int t = blockIdx.x * blockDim.x + threadIdx.x;
  int i = t >> LOGH, h = t & ((1 << LOGH) - 1);
  if (i >= Etot) return;
  int s = (i < E) ? src[i] : (i - E);
  int d = (i < E) ? dst[i] : (i - E);
  const int H = 1 << LOGH;
  float e = es[s * H + h] + ed[d * H + h];
  e = e > 0.f ? e : NEG_SLOPE * e;
  float alpha = __expf(e - mx[d * H + h]) / ssum[d * H + h];
  const float* hp = hsrc + ((size_t)s * H + h) * C;
  float* ap = agg + ((size_t)d * H + h) * C;
#pragma unroll
  for (int c = 0; c < C; ++c) atomicAdd(&ap[c], alpha * hp[c]);
}

// ---------------- bias + ELU (in place) -------------------------------------
__global__ void bias_elu_k(float* __restrict__ a, const float* __restrict__ b, int total) {
  int i = blockIdx.x * blockDim.x + threadIdx.x;
  if (i >= total) return;
  float v = a[i] + b[i & (HID - 1)];
  a[i] = v > 0.f ? v : (__expf(v) - 1.f);
}

// ---------------- GEMM2 (64->16 per node) + layer-2 attention logits --------
__global__ void __launch_bounds__(256) gemm2_node(
    const float* __restrict__ hin, const float* __restrict__ W2,
    const float* __restrict__ asrc, const float* __restrict__ adst,
    float* __restrict__ h2, float* __restrict__ es2, float* __restrict__ ed2,
    int nnodes) {
  __shared__ float w[HID * NCLS];
  __shared__ float as[NCLS], ad[NCLS];
  for (int i = threadIdx.x; i < HID * NCLS; i += blockDim.x) w[i] = W2[i];
  if (threadIdx.x < NCLS) { as[threadIdx.x] = asrc[threadIdx.x]; ad[threadIdx.x] = adst[threadIdx.x]; }
  __syncthreads();
  int nid = blockIdx.x * blockDim.x + threadIdx.x;
  if (nid >= nnodes) return;
  const float* hr = hin + (size_t)nid * HID;
  float out[NCLS];
#pragma unroll
  for (int j = 0; j < NCLS; ++j) out[j] = 0.f;
  for (int k = 0; k < HID; ++k) {
    float hv = hr[k];
#pragma unroll
    for (int j = 0; j < NCLS; ++j) out[j] += hv * w[k * NCLS + j];
  }
  float e_s = 0.f, e_d = 0.f;
  float* op = h2 + (size_t)nid * NCLS;
#pragma unroll
  for (int j = 0; j < NCLS; ++j) {
    op[j] = out[j];
    e_s += out[j] * as[j];
    e_d += out[j] * ad[j];
  }
  es2[nid] = e_s; ed2[nid] = e_d;
}

// ---------------- bias + log_softmax ----------------------------------------
__global__ void final_k(const float* __restrict__ agg2, const float* __restrict__ b2,
                        float* __restrict__ out, int nnodes) {
  int nid = blockIdx.x * blockDim.x + threadIdx.x;
  if (nid >= nnodes) return;
  const float* ap = agg2 + (size_t)nid * NCLS;
  float v[NCLS], mx = -INFINITY;
#pragma unroll
  for (int j = 0; j < NCLS; ++j) { v[j] = ap[j] + b2[j]; mx = fmaxf(mx, v[j]); }
  float ssum = 0.f;
#pragma unroll
  for (int j = 0; j < NCLS; ++j) ssum += __expf(v[j] - mx);
  float lse = mx + logf(ssum);
  float* op = out + (size_t)nid * NCLS;
#pragma unroll
  for (int j = 0; j < NCLS; ++j) op[j] = v[j] - lse;
}

// ============================================================================
extern "C" void kernel_launch(void* const* d_in, const int* in_sizes, int n_in,
                              void* d_out, int out_size, void* d_ws, size_t ws_size,
                              hipStream_t stream) {
  const float* x   = (const float*)d_in[0];
  const int*   ei  = (const int*)d_in[1];
  const float* W1  = (const float*)d_in[2];
  const float* as1 = (const float*)d_in[3];
  const float* ad1 = (const float*)d_in[4];
  const float* b1  = (const float*)d_in[5];
  const float* W2  = (const float*)d_in[6];
  const float* as2 = (const float*)d_in[7];
  const float* ad2 = (const float*)d_in[8];
  const float* b2  = (const float*)d_in[9];
  float* out = (float*)d_out;

  const int Nn = in_sizes[0] / FIN;     // 100000
  const int E  = in_sizes[1] / 2;       // 3200000
  const int* src = ei;
  const int* dstp = ei + E;
  const int Etot = E + Nn;              // + self loops

  float* ws   = (float*)d_ws;
  float* h1   = ws; ws += (size_t)Nn * HID;
  float* es1  = ws; ws += (size_t)Nn * 8;
  float* ed1  = ws; ws += (size_t)Nn * 8;
  float* m1   = ws; ws += (size_t)Nn * 8;
  float* s1   = ws; ws += (size_t)Nn * 8;
  float* agg1 = ws; ws += (size_t)Nn * HID;
  float* h2   = ws; ws += (size_t)Nn * NCLS;
  float* es2  = ws; ws += (size_t)Nn;
  float* ed2  = ws; ws += (size_t)Nn;
  float* m2   = ws; ws += (size_t)Nn;
  float* s2   = ws; ws += (size_t)Nn;
  float* agg2 = ws; ws += (size_t)Nn * NCLS;

  // ---- layer 1 ----
  int tiles = (Nn + 15) / 16;
  gemm1_wmma<<<(tiles + 3) / 4, 128, 0, stream>>>(x, W1, h1, Nn);

  int nH = Nn * 8;
  node_attn1<<<(nH + 255) / 256, 256, 0, stream>>>(h1, as1, ad1, es1, ed1, nH);

  size_t a1n = (size_t)Nn * HID;
  fillf<<<(nH + 255) / 256, 256, 0, stream>>>(m1, -INFINITY, (size_t)nH);
  fillf<<<(nH + 255) / 256, 256, 0, stream>>>(s1, 0.f, (size_t)nH);
  fillf<<<(int)((a1n + 255) / 256), 256, 0, stream>>>(agg1, 0.f, a1n);

  long long t1 = (long long)Etot * 8;
  int g1 = (int)((t1 + 255) / 256);
  edge_max_k<3><<<g1, 256, 0, stream>>>(src, dstp, E, Etot, es1, ed1, m1);
  edge_sum_k<3><<<g1, 256, 0, stream>>>(src, dstp, E, Etot, es1, ed1, m1, s1);
  edge_aggr_k<3, 8><<<g1, 256, 0, stream>>>(src, dstp, E, Etot, es1, ed1, m1, s1, h1, agg1);

  bias_elu_k<<<(int)((a1n + 255) / 256), 256, 0, stream>>>(agg1, b1, (int)a1n);

  // ---- layer 2 ----
  gemm2_node<<<(Nn + 255) / 256, 256, 0, stream>>>(agg1, W2, as2, ad2, h2, es2, ed2, Nn);

  fillf<<<(Nn + 255) / 256, 256, 0, stream>>>(m2, -INFINITY, (size_t)Nn);
  fillf<<<(Nn + 255) / 256, 256, 0, stream>>>(s2, 0.f, (size_t)Nn);
  fillf<<<(Nn * NCLS + 255) / 256, 256, 0, stream>>>(agg2, 0.f, (size_t)Nn * NCLS);

  int g2 = (Etot + 255) / 256;
  edge_max_k<0><<<g2, 256, 0, stream>>>(src, dstp, E, Etot, es2, ed2, m2);
  edge_sum_k<0><<<g2, 256, 0, stream>>>(src, dstp, E, Etot, es2, ed2, m2, s2);
  edge_aggr_k<0, 16><<<g2, 256, 0, stream>>>(src, dstp, E, Etot, es2, ed2, m2, s2, h2, agg2);

  final_k<<<(Nn + 255) / 256, 256, 0, stream>>>(agg2, b2, out, Nn);
}